// L1_13494787244126
// MI455X (gfx1250) — compile-verified
//
#include <hip/hip_runtime.h>
#include <stdint.h>

#define TILE    64
#define DDIM    256
#define MDIM    8192
#define NDIM    8192
#define RSTRIDE 260   // floats per LDS row: 1040B = 16B-aligned, staggers banks by 4/row

// Low 32 bits of a flat pointer into the LDS aperture are the LDS byte address
// (ISA 10.2); hardware adds the wave's LDS_BASE.
__device__ __forceinline__ unsigned lds_off(const void* p) {
  return (unsigned)(uintptr_t)p;
}

// gfx1250 async global->LDS copy, 16B per lane, tracked on ASYNCcnt.
__device__ __forceinline__ void async_load_b128(unsigned lds, const void* gaddr) {
  asm volatile("global_load_async_to_lds_b128 %0, %1, off"
               :: "v"(lds), "v"(gaddr) : "memory");
}

__device__ __forceinline__ void wait_async_zero() {
  asm volatile("s_wait_asynccnt 0" ::: "memory");
}

__global__ __launch_bounds__(256, 2)
void neg_l1_tile_kernel(const float* __restrict__ x,
                        const float* __restrict__ y,
                        float* __restrict__ out) {
  __shared__ __align__(16) float sx[TILE * RSTRIDE];
  __shared__ __align__(16) float sy[TILE * RSTRIDE];

  const int  tid = (int)threadIdx.x;
  const long i0  = (long)blockIdx.y * TILE;
  const long j0  = (long)blockIdx.x * TILE;

  // ---- Stage x[i0:i0+64, :] and y[j0:j0+64, :] into LDS via async B128 ----
  {
    const int r = tid >> 6;                       // row within a 4-row group
    const int c = tid & 63;                       // 16B chunk within a 1KB row
    const float* gx = x + (i0 + r) * DDIM + c * 4;
    const float* gy = y + (j0 + r) * DDIM + c * 4;
    unsigned lx = lds_off(&sx[r * RSTRIDE + c * 4]);
    unsigned ly = lds_off(&sy[r * RSTRIDE + c * 4]);
#pragma unroll
    for (int q = 0; q < 16; ++q) {                // 16 x 4-row groups = 64 rows
      async_load_b128(lx, gx);
      async_load_b128(ly, gy);
      gx += 4 * DDIM;
      gy += 4 * DDIM;
      lx += 4 * RSTRIDE * 4;
      ly += 4 * RSTRIDE * 4;
    }
  }
  wait_async_zero();
  __syncthreads();

  const int tx = tid & 15;   // j-group: this thread owns j = j0 + tx + 16*c
  const int ty = tid >> 4;   // i-group: this thread owns i = i0 + ty*4 + r

  float acc[4][4];
#pragma unroll
  for (int r = 0; r < 4; ++r)
#pragma unroll
    for (int c = 0; c < 4; ++c)
      acc[r][c] = 0.0f;

  const float4* xrow[4];
  const float4* yrow[4];
#pragma unroll
  for (int r = 0; r < 4; ++r)
    xrow[r] = (const float4*)&sx[(ty * 4 + r) * RSTRIDE];
#pragma unroll
  for (int c = 0; c < 4; ++c)
    yrow[c] = (const float4*)&sy[(tx + 16 * c) * RSTRIDE];

  // Main loop: per 4-k step, 8 ds_load_b128 + 128 VALU (v_sub + v_sub with |abs|).
#pragma unroll 2
  for (int k4 = 0; k4 < DDIM / 4; ++k4) {
    float4 xv[4], yv[4];
#pragma unroll
    for (int r = 0; r < 4; ++r) xv[r] = xrow[r][k4];
#pragma unroll
    for (int c = 0; c < 4; ++c) yv[c] = yrow[c][k4];
#pragma unroll
    for (int r = 0; r < 4; ++r)
#pragma unroll
      for (int c = 0; c < 4; ++c) {
        acc[r][c] -= __builtin_fabsf(xv[r].x - yv[c].x);
        acc[r][c] -= __builtin_fabsf(xv[r].y - yv[c].y);
        acc[r][c] -= __builtin_fabsf(xv[r].z - yv[c].z);
        acc[r][c] -= __builtin_fabsf(xv[r].w - yv[c].w);
      }
  }

  // Accumulated negative sums directly; coalesced b32 stores (16 lanes cover 64B).
#pragma unroll
  for (int r = 0; r < 4; ++r) {
    float* orow = out + (i0 + ty * 4 + r) * (long)MDIM + j0;
#pragma unroll
    for (int c = 0; c < 4; ++c)
      orow[tx + 16 * c] = acc[r][c];
  }
}

extern "C" void kernel_launch(void* const* d_in, const int* in_sizes, int n_in,
                              void* d_out, int out_size, void* d_ws, size_t ws_size,
                              hipStream_t stream) {
  (void)in_sizes; (void)n_in; (void)out_size; (void)d_ws; (void)ws_size;
  const float* x = (const float*)d_in[0];
  const float* y = (const float*)d_in[1];
  // d_in[2] = chunk_size (reference-internal chunking knob) — unused.
  float* out = (float*)d_out;

  dim3 grid(MDIM / TILE, NDIM / TILE);  // (j-tiles, i-tiles) = 128 x 128
  neg_l1_tile_kernel<<<grid, dim3(256), 0, stream>>>(x, y, out);
}